// TopicSegment_37830071943311
// MI455X (gfx1250) — compile-verified
//
#include <hip/hip_runtime.h>

typedef float v2f __attribute__((ext_vector_type(2)));
typedef float v4f __attribute__((ext_vector_type(4)));
typedef float v8f __attribute__((ext_vector_type(8)));

#define B_DIM 8
#define S_DIM 4096
#define D_DIM 1024
#define NEG_BIG (-1.0e18f)

// stable log(exp(a)+exp(b))
__device__ __forceinline__ float lse2(float a, float b) {
    float mx = fmaxf(a, b);
    return mx + log1pf(__expf(-fabsf(a - b)));
}

// ---------------------------------------------------------------------------
// Kernel 1: logits[(row),n] = sum_k x[row][k] * W[n][k] + b[n], n in {0,1}
// One wave per 16-row tile. fp32 WMMA 16x16x4: A = x tile, B = W^T chunk
// (columns >= 2 are zeroed via a lane mask, keeping EXEC all-1s for WMMA).
// K-chunks are remapped so each lane does contiguous b128 loads:
//   lane half h, iteration i: loads x[row][8i+4h .. 8i+4h+3] (16B aligned)
//   wmma #0 consumes (.x,.y), wmma #1 consumes (.z,.w); B loaded identically.
// ---------------------------------------------------------------------------
__global__ __launch_bounds__(256) void topic_logits_wmma(
    const float* __restrict__ x, const float* __restrict__ W,
    const float* __restrict__ bias, float* __restrict__ logits) {
    const int lane = threadIdx.x & 31;
    const int wave = blockIdx.x * (blockDim.x >> 5) + (threadIdx.x >> 5);
    const int h = lane >> 4;   // which half of the wave (K sub-pair)
    const int l = lane & 15;   // A row index / B column index
    const long row0 = (long)wave * 16;

    const float* arow = x + (row0 + l) * D_DIM + 4 * h;
    const float* wrow = W + (l & 1) * D_DIM + 4 * h;
    const float wmask = (l < 2) ? 1.0f : 0.0f;   // B columns 2..15 are zero

    v8f c = {0.0f, 0.0f, 0.0f, 0.0f, 0.0f, 0.0f, 0.0f, 0.0f};
#pragma unroll 4
    for (int i = 0; i < D_DIM / 8; ++i) {
        v4f a4 = *(const v4f*)(arow + 8 * i);
        v4f w4 = *(const v4f*)(wrow + 8 * i);
        v2f a0 = {a4.x, a4.y};
        v2f b0 = {w4.x * wmask, w4.y * wmask};
        c = __builtin_amdgcn_wmma_f32_16x16x4_f32(false, a0, false, b0,
                                                  (short)0, c, false, false);
        v2f a1 = {a4.z, a4.w};
        v2f b1 = {w4.z * wmask, w4.w * wmask};
        c = __builtin_amdgcn_wmma_f32_16x16x4_f32(false, a1, false, b1,
                                                  (short)0, c, false, false);
    }

    // C/D layout: VGPR r -> M = r (lanes 0-15) or M = 8+r (lanes 16-31), N = l
    if (l < 2) {
        const float bb = bias[l];
        const long mrow = row0 + 8 * h;
#pragma unroll
        for (int r = 0; r < 8; ++r)
            logits[(mrow + r) * 2 + l] = c[r] + bb;
    }
}

// ---------------------------------------------------------------------------
// Kernel 2: CRF neg-log-likelihood as a parallel ordered reduction.
// Forward recurrence alpha_t = alpha_{t-1} (x) M_t in the log-semiring,
// M_t[i][j] = T[i][j] + e_t[j]. Only logZ is needed -> ordered reduction of
// 2x2 log-matrices. 1024 threads: 128 per batch, 32 matrices folded locally,
// then a 7-level order-preserving pairwise tree in LDS. Numerator terms and
// mask counts reduce alongside. Masked-off steps contribute identity (skip).
// ---------------------------------------------------------------------------
__global__ __launch_bounds__(1024) void topic_crf_reduce(
    const float* __restrict__ logits, const int* __restrict__ labels,
    const unsigned char* __restrict__ mask,
    const float* __restrict__ startT, const float* __restrict__ endT,
    const float* __restrict__ trans, float* __restrict__ out) {
    __shared__ float s00[1024], s01[1024], s10[1024], s11[1024];
    __shared__ float snum[1024];
    __shared__ int scnt[1024];
    __shared__ float sllh[B_DIM];

    const int tid = threadIdx.x;
    const int b = tid >> 7;        // batch 0..7
    const int i = tid & 127;       // slot within batch
    const int CH = S_DIM / 128;    // 32 timesteps per slot

    const float* lg = logits + (size_t)b * S_DIM * 2;
    const int* lb = labels + b * S_DIM;
    const unsigned char* mk = mask + b * S_DIM;
    const float t00 = trans[0], t01 = trans[1], t10 = trans[2], t11 = trans[3];

    // local fold over t in [t0, t1)
    float P00 = 0.0f, P01 = NEG_BIG, P10 = NEG_BIG, P11 = 0.0f;  // identity
    float numloc = 0.0f;
    int cntloc = (i == 0) ? (mk[0] ? 1 : 0) : 0;  // count t=0
    const int t0 = (i == 0) ? 1 : i * CH;         // t=0 handled separately
    const int t1 = (i + 1) * CH > S_DIM ? S_DIM : (i + 1) * CH;
    for (int t = t0; t < t1; ++t) {
        const float e0 = lg[2 * t];
        const float e1 = lg[2 * t + 1];
        const int y = lb[t];
        const int yp = lb[t - 1];
        if (mk[t]) {
            numloc += ((y == 0) ? e0 : e1) + trans[yp * 2 + y];
            cntloc += 1;
            const float m00 = t00 + e0, m01 = t01 + e1;
            const float m10 = t10 + e0, m11 = t11 + e1;
            const float n00 = lse2(P00 + m00, P01 + m10);
            const float n01 = lse2(P00 + m01, P01 + m11);
            const float n10 = lse2(P10 + m00, P11 + m10);
            const float n11 = lse2(P10 + m01, P11 + m11);
            P00 = n00; P01 = n01; P10 = n10; P11 = n11;
        }
    }
    s00[tid] = P00; s01[tid] = P01; s10[tid] = P10; s11[tid] = P11;
    snum[tid] = numloc; scnt[tid] = cntloc;
    __syncthreads();

    // order-preserving pairwise tree: left operand is the earlier chunk
    for (int s = 1; s < 128; s <<= 1) {
        if ((i & (2 * s - 1)) == 0) {
            const int j = tid + s;
            const float a00 = s00[tid], a01 = s01[tid];
            const float a10 = s10[tid], a11 = s11[tid];
            const float b00 = s00[j], b01 = s01[j];
            const float b10 = s10[j], b11 = s11[j];
            s00[tid] = lse2(a00 + b00, a01 + b10);
            s01[tid] = lse2(a00 + b01, a01 + b11);
            s10[tid] = lse2(a10 + b00, a11 + b10);
            s11[tid] = lse2(a10 + b01, a11 + b11);
            snum[tid] += snum[j];
            scnt[tid] += scnt[j];
        }
        __syncthreads();
    }

    if (i == 0) {
        const int y0 = lb[0];
        const int cnt = scnt[tid];
        const int last = lb[cnt - 1];
        float num = startT[y0] + ((y0 == 0) ? lg[0] : lg[1]) + snum[tid] +
                    endT[last];
        const float a0 = startT[0] + lg[0];
        const float a1 = startT[1] + lg[1];
        const float r0 = lse2(a0 + s00[tid], a1 + s10[tid]);
        const float r1 = lse2(a0 + s01[tid], a1 + s11[tid]);
        const float logZ = lse2(r0 + endT[0], r1 + endT[1]);
        sllh[b] = num - logZ;
    }
    __syncthreads();
    if (tid == 0) {
        float s = 0.0f;
#pragma unroll
        for (int k = 0; k < B_DIM; ++k) s += sllh[k];
        out[0] = -s;   // reduction='sum', module returns -llh
    }
}

extern "C" void kernel_launch(void* const* d_in, const int* in_sizes, int n_in,
                              void* d_out, int out_size, void* d_ws, size_t ws_size,
                              hipStream_t stream) {
    (void)in_sizes; (void)n_in; (void)out_size; (void)d_ws; (void)ws_size;
    const float* x = (const float*)d_in[0];
    const int* labels = (const int*)d_in[1];
    const unsigned char* mask = (const unsigned char*)d_in[2];
    const float* W = (const float*)d_in[3];
    const float* bias = (const float*)d_in[4];
    const float* startT = (const float*)d_in[5];
    const float* endT = (const float*)d_in[6];
    const float* trans = (const float*)d_in[7];

    float* out = (float*)d_out;       // out[0] = neg log-likelihood
    float* logits = out + 1;          // out[1..] = logits (B,S,2)

    // (B*S)/16 = 2048 row tiles, 8 waves per block -> 256 blocks
    topic_logits_wmma<<<256, 256, 0, stream>>>(x, W, bias, logits);
    topic_crf_reduce<<<1, 1024, 0, stream>>>(logits, labels, mask, startT,
                                             endT, trans, out);
}